// TianCNN3_10204842295638
// MI455X (gfx1250) — compile-verified
//
#include <hip/hip_runtime.h>
#include <hip/hip_bf16.h>

// B=8192, T=100, X=256, window 30:100 (70 samples), K=50 -> 21 conv outputs,
// relu+max over time -> feat[8192,256]; then feat @ fc_w^T -> out[8192,300].

#define BATCH   8192
#define TDIM    100
#define XDIM    256
#define WIN     70
#define KSZ     50
#define JOUT    21      // WIN - KSZ + 1
#define NOUT    300     // Y_DIM * 50
#define NTILES  19      // ceil(300/16)

typedef float v2f __attribute__((ext_vector_type(2)));
typedef float v8f __attribute__((ext_vector_type(8)));

// ---------------- Stage 1: fused slice + depthwise conv + relu + timemax ----
// grid = (BATCH), block = (256).  thread c handles channel c of batch b.
// Coalesced: lanes 0..255 read contiguous 1024B per time-row.
__global__ void __launch_bounds__(256)
conv_relu_max_kernel(const float* __restrict__ seq,
                     const float* __restrict__ conv_w,
                     float* __restrict__ feat) {
    const int c = threadIdx.x;
    const int b = blockIdx.x;

    // per-channel weights (reused 70x from registers; conv_w is L2-resident)
    float w[KSZ];
#pragma unroll
    for (int k = 0; k < KSZ; ++k) w[k] = conv_w[c * KSZ + k];

    float acc[JOUT];
#pragma unroll
    for (int j = 0; j < JOUT; ++j) acc[j] = 0.0f;

    // stream the 70-sample window once, non-temporal (587MB total stream;
    // don't pollute the 192MB L2 that holds feat/fc_w for stage 2)
    const float* xp = seq + ((size_t)b * TDIM + 30) * XDIM + c;
#pragma unroll
    for (int i = 0; i < WIN; ++i) {
        float xi = __builtin_nontemporal_load(xp + (size_t)i * XDIM);
        const int jlo = (i - (KSZ - 1)) > 0 ? (i - (KSZ - 1)) : 0;
        const int jhi = i < (JOUT - 1) ? i : (JOUT - 1);
#pragma unroll
        for (int j = jlo; j <= jhi; ++j) {
            acc[j] = fmaf(xi, w[i - j], acc[j]);
        }
    }

    // relu then max over time == max(0, max_j acc[j])
    float m = 0.0f;
#pragma unroll
    for (int j = 0; j < JOUT; ++j) m = fmaxf(m, acc[j]);

    feat[(size_t)b * XDIM + c] = m;
}

// ---------------- Stage 2: feat[8192,256] @ fc_w^T[256,300] via WMMA f32 ----
// One wave per 16x16 output tile. V_WMMA_F32_16X16X4_F32, K-loop of 64.
// A lane layout (16x4 f32): a[v] = A[lane%16][2*(lane/16)+v]
// B lane layout (4x16 f32): b[v] = B[2*(lane/16)+v][lane%16] = fc_w[n][k]
// C/D layout: d[v] = D[8*(lane/16)+v][lane%16]
__global__ void __launch_bounds__(256)
fc_wmma_kernel(const float* __restrict__ feat,
               const float* __restrict__ fcw,
               float* __restrict__ out) {
    const int lane = threadIdx.x & 31;
    const int wave = (blockIdx.x * blockDim.x + threadIdx.x) >> 5;

    const int mt = wave / NTILES;       // 0..511
    const int nt = wave - mt * NTILES;  // 0..18
    const int m0 = mt * 16;
    const int n0 = nt * 16;

    const int l16  = lane & 15;
    const int half = lane >> 4;         // 0 or 1
    const int kb   = half * 2;          // K sub-offset for this lane

    const int rowA = m0 + l16;
    int rowB = n0 + l16;
    if (rowB > NOUT - 1) rowB = NOUT - 1;   // clamp (branch-free wrt EXEC);
                                            // clamped columns are never stored

    const float* ap = feat + (size_t)rowA * XDIM + kb;
    const float* bp = fcw  + (size_t)rowB * XDIM + kb;

    v8f acc = {};
#pragma unroll
    for (int k0 = 0; k0 < XDIM; k0 += 4) {
        v2f a = *(const v2f*)(ap + k0);   // global_load_b64
        v2f b = *(const v2f*)(bp + k0);   // global_load_b64
        // (neg_a, A, neg_b, B, c_mod, C, reuse_a, reuse_b)
        acc = __builtin_amdgcn_wmma_f32_16x16x4_f32(
            false, a, false, b, (short)0, acc, false, false);
    }

    const int col   = n0 + l16;
    const int rbase = m0 + 8 * half;
    if (col < NOUT) {
#pragma unroll
        for (int v = 0; v < 8; ++v) {
            out[(size_t)(rbase + v) * NOUT + col] = acc[v];
        }
    }
}

extern "C" void kernel_launch(void* const* d_in, const int* in_sizes, int n_in,
                              void* d_out, int out_size, void* d_ws, size_t ws_size,
                              hipStream_t stream) {
    const float* seq    = (const float*)d_in[0];   // [8192,100,256] f32
    const float* conv_w = (const float*)d_in[1];   // [256,1,50]     f32
    const float* fc_w   = (const float*)d_in[2];   // [300,256]      f32
    float*       out    = (float*)d_out;           // [8192,300]     f32
    float*       feat   = (float*)d_ws;            // [8192,256] f32 = 8.4 MB scratch

    conv_relu_max_kernel<<<dim3(BATCH), dim3(256), 0, stream>>>(seq, conv_w, feat);

    const int tiles  = (BATCH / 16) * NTILES;      // 512 * 19 = 9728 waves
    const int blocks = tiles / 8;                  // 8 waves (256 thr) per block = 1216
    fc_wmma_kernel<<<dim3(blocks), dim3(256), 0, stream>>>(feat, fc_w, out);
}